// PositionalEncoding_80590766342357
// MI455X (gfx1250) — compile-verified
//
#include <hip/hip_runtime.h>

// CDNA5 / gfx1250: wave32, WMMA 16x16 shapes.
typedef __attribute__((ext_vector_type(2))) float v2f;   // A/B of V_WMMA_F32_16X16X4_F32 (16x4 / 4x16 f32 = 2 VGPRs)
typedef __attribute__((ext_vector_type(8))) float v8f;   // 16x16 f32 C/D accumulator (8 VGPRs)

#define DM        1024
#define RESF      200.0f
// log2(200)/1024
#define L2RES_OVER_D 0.00746470331032697f
#define PI_OVER_2 1.57079632679489662f

// Each block: 256 threads = 8 waves. Wave w owns channels [w*128, w*128+128) as 8
// 16-wide channel tiles. Block covers 64 positions as 4 position tiles of 16.
// Per (pos-tile, ch-tile): enc(16x16) = A_pos x B_pos + A_neg x B_neg via two
// accumulating v_wmma_f32_16x16x4_f32 rank-1 updates (only K=0 populated), then
// out = sin(enc + phase) with phase = (ch odd ? pi/2 : 0)  [cos == shifted sin].
__global__ __launch_bounds__(256) void pe_wmma_kernel(const float* __restrict__ x,
                                                      float* __restrict__ out,
                                                      int npos) {
    const int lane   = threadIdx.x & 31;
    const int wave   = threadIdx.x >> 5;
    const int n      = lane & 15;   // column within 16x16 tile = channel offset
    const int hi     = lane >> 4;   // 0: rows M=v ; 1: rows M=v+8 (C/D layout)
    const bool lo    = (hi == 0);
    const int chbase = wave * 128;

    // ---- Hoisted per-channel scale tiles (B matrices) and sin/cos phase ----
    v2f   Bpos[8], Bneg[8];
    float phase[8];
#pragma unroll
    for (int t = 0; t < 8; ++t) {
        const int ch  = chbase + t * 16 + n;
        const int chn = (DM - 1) - ch;                       // flipped index (x < 0 branch)
        const float spos = __builtin_exp2f(-(float)(ch  & ~1) * L2RES_OVER_D);  // RES^(-2*(ch//2)/D)
        const float sneg = __builtin_exp2f(-(float)(chn & ~1) * L2RES_OVER_D);
        v2f bp; bp.x = lo ? spos : 0.0f; bp.y = 0.0f;        // B[K=0, N=n]; K=1..3 zero
        v2f bn; bn.x = lo ? sneg : 0.0f; bn.y = 0.0f;
        Bpos[t] = bp;
        Bneg[t] = bn;
        phase[t] = (ch & 1) ? PI_OVER_2 : 0.0f;              // odd channel -> cos = sin(.+pi/2)
    }

    // ---- 4 position tiles of 16 per block ----
    for (int pt = 0; pt < 4; ++pt) {
        const int posbase = blockIdx.x * 64 + pt * 16;
        if (posbase >= npos) break;

        float xv = 0.0f;
        if (lo) {
            const int p = posbase + n;                       // lanes 0..15 load 16 positions
            xv = (p < npos) ? x[p] : 0.0f;
        }
        xv = fminf(fmaxf(xv, -RESF), RESF);                  // clamp to [-RES, RES]

        v2f Apos, Aneg;                                      // A[M=n, K=0]; K=1..3 zero
        Apos.x = (lo && xv >= 0.0f) ?  xv : 0.0f; Apos.y = 0.0f;
        Aneg.x = (lo && xv <  0.0f) ? -xv : 0.0f; Aneg.y = 0.0f;

#pragma unroll
        for (int t = 0; t < 8; ++t) {
            v8f acc = {};
            // enc = A_pos x B_pos
            acc = __builtin_amdgcn_wmma_f32_16x16x4_f32(false, Apos, false, Bpos[t],
                                                        (short)0, acc, false, false);
            // enc += A_neg x B_neg
            acc = __builtin_amdgcn_wmma_f32_16x16x4_f32(false, Aneg, false, Bneg[t],
                                                        (short)0, acc, false, false);

            // D layout: lane -> (N = n, rows M = v + 8*hi). Lanes 0-15 / 16-31 each
            // store a contiguous 64B run per instruction -> coalesced.
            const int    ch   = chbase + t * 16 + n;
            const size_t base = (size_t)(posbase + hi * 8) * DM + ch;
#pragma unroll
            for (int v = 0; v < 8; ++v) {
                out[base + (size_t)v * DM] = __sinf(acc[v] + phase[t]);
            }
        }
    }
}

extern "C" void kernel_launch(void* const* d_in, const int* in_sizes, int n_in,
                              void* d_out, int out_size, void* d_ws, size_t ws_size,
                              hipStream_t stream) {
    const float* x = (const float*)d_in[0];
    float* out     = (float*)d_out;
    const int npos = in_sizes[0];                 // 4*512*32 = 65536 scalar positions
    const int blocks = (npos + 63) / 64;          // 64 positions per block
    pe_wmma_kernel<<<blocks, 256, 0, stream>>>(x, out, npos);
}